// Coo2Cel_28278064676904
// MI455X (gfx1250) — compile-verified
//
#include <hip/hip_runtime.h>
#include <stdint.h>

// ---------------------------------------------------------------------------
// Coo2Cel neighbor math for MI455X (gfx1250, wave32).
//
// Roofline: output = B*N*N*5 floats = 671 MB write-once stream -> ~29 us at
// 23.3 TB/s HBM. Compute ~1.3 GFLOP is comparable in wave-instruction count,
// so: precompute frac once (WMMA f32 16x16x4 — the only matmul-shaped step;
// the per-pair min-image round() is nonlinear and cannot be a matmul), then a
// store-bound streaming pair kernel with non-temporal b128 stores (output
// >> 192MB L2, never re-read).
// ---------------------------------------------------------------------------

typedef float v2f  __attribute__((ext_vector_type(2)));
typedef float v4f  __attribute__((ext_vector_type(4)));
typedef float v8f  __attribute__((ext_vector_type(8)));

#define RC2 36.0f

// --- Kernel 0: per-batch 3x3 inverse of cel_mat (cofactor method) -----------
__global__ void inv3x3_kernel(const float* __restrict__ cel,
                              float* __restrict__ inv, int B) {
    int b = threadIdx.x;
    if (b >= B) return;
    const float* C = cel + b * 9;
    float a = C[0], bb = C[1], c = C[2];
    float d = C[3], e  = C[4], f = C[5];
    float g = C[6], h  = C[7], i = C[8];
    float A00 =  (e * i - f * h);
    float A01 = -(d * i - f * g);
    float A02 =  (d * h - e * g);
    float A10 = -(bb * i - c * h);
    float A11 =  (a * i - c * g);
    float A12 = -(a * h - bb * g);
    float A20 =  (bb * f - c * e);
    float A21 = -(a * f - c * d);
    float A22 =  (a * e - bb * d);
    float det = a * A00 + bb * A01 + c * A02;
    float r = 1.0f / det;
    float* M = inv + b * 9;
    // inv[i][j] = cof(j,i)/det
    M[0] = A00 * r; M[1] = A10 * r; M[2] = A20 * r;
    M[3] = A01 * r; M[4] = A11 * r; M[5] = A21 * r;
    M[6] = A02 * r; M[7] = A12 * r; M[8] = A22 * r;
}

// --- Kernel 1: frac = pos * inv(cel), 16 atoms per wave via f32 WMMA --------
// A (16x4 f32, ISA 7.12.2): lanes 0-15 hold K=0 (V0), K=1 (V1); lanes 16-31
// hold K=2 (V0), K=3 (V1). M = lane%16 = atom-in-tile.
// B (4x16 f32, mirrored K grouping): V0 = row K0 (lo lanes) / K2 (hi lanes),
// V1 = row K1 (lo) / K3 (hi); N = lane%16 = frac component (cols 3..15 = 0).
// D (16x16 f32): VGPR v, lanes 0-15 -> M=v, N=lane; lanes 16-31 -> M=v+8.
__global__ void frac_wmma_kernel(const float* __restrict__ pos,   // [B*N,3]
                                 const float* __restrict__ inv,   // [B,9]
                                 float* __restrict__ frac,        // [B*N,3]
                                 int N, int total) {
    const int lane = threadIdx.x & 31;
    const int wave = threadIdx.x >> 5;
    const int tile = blockIdx.x * (blockDim.x >> 5) + wave;
    const int base = tile * 16;
    if (base >= total) return;                 // wave-uniform: EXEC stays all-1
    const int  m  = lane & 15;
    const bool hi = lane >= 16;

    int atom = base + m;
    if (atom >= total) atom = total - 1;       // clamp loads, guard stores below
    const float* P = pos + (size_t)atom * 3;
    float px = P[0], py = P[1], pz = P[2];

    v2f a;
    a.x = hi ? pz : px;
    a.y = hi ? 0.0f : py;

    const int batch = base / N;                // tiles don't straddle (N%16==0)
    const float* M = inv + batch * 9;
    const int n = m;                           // D column = frac component
    float b0 = 0.0f, b1 = 0.0f;
    if (n < 3) {                               // select-based, reconverges
        b0 = hi ? M[2 * 3 + n] : M[0 * 3 + n];
        b1 = hi ? 0.0f         : M[1 * 3 + n];
    }
    v2f bm; bm.x = b0; bm.y = b1;

    v8f acc = {};
#if defined(__gfx1250__) && __has_builtin(__builtin_amdgcn_wmma_f32_16x16x4_f32)
    v8f dmat = __builtin_amdgcn_wmma_f32_16x16x4_f32(
        false, a, false, bm, (short)0, acc, false, false);
    if (n < 3) {
#pragma unroll
        for (int v = 0; v < 8; ++v) {
            int row = base + v + (hi ? 8 : 0);
            if (row < total) frac[(size_t)row * 3 + n] = dmat[v];
        }
    }
#else
    (void)a; (void)bm; (void)acc;
    // VALU fallback: each lane 0-15 computes its own atom's frac row.
    if (!hi && base + m < total) {
        const float* Mi = inv + batch * 9;
        float f0 = px * Mi[0] + py * Mi[3] + pz * Mi[6];
        float f1 = px * Mi[1] + py * Mi[4] + pz * Mi[7];
        float f2 = px * Mi[2] + py * Mi[5] + pz * Mi[8];
        float* F = frac + (size_t)(base + m) * 3;
        F[0] = f0; F[1] = f1; F[2] = f2;
    }
#endif
}

// --- Kernel 2: streaming pair kernel (store-bound) --------------------------
// 1 thread = 4 consecutive j. Non-temporal b128 stores; fmaf/rintf ->
// v_fma_f32 / v_rndne_f32 (matches jnp.round round-half-even).
__global__ void pair_kernel(const float* __restrict__ frac,   // [B*N,3]
                            const float* __restrict__ cel,    // [B,3,3]
                            const uint8_t* __restrict__ pbc,  // [B,3] bool
                            const uint8_t* __restrict__ ent,  // [B,N] bool
                            float* __restrict__ ovec,         // [B,N,N,3]
                            float* __restrict__ osod,         // [B,N,N]
                            float* __restrict__ omsk,         // [B,N,N]
                            int N) {
    const int b  = blockIdx.z;
    const int i  = blockIdx.y;
    const int j0 = (blockIdx.x * blockDim.x + threadIdx.x) * 4;
    if (j0 >= N) return;

    const float* C = cel + b * 9;
    const float c00 = C[0], c01 = C[1], c02 = C[2];
    const float c10 = C[3], c11 = C[4], c12 = C[5];
    const float c20 = C[6], c21 = C[7], c22 = C[8];
    const uint8_t* pb = pbc + b * 3;
    const float p0 = pb[0] ? 1.0f : 0.0f;
    const float p1 = pb[1] ? 1.0f : 0.0f;
    const float p2 = pb[2] ? 1.0f : 0.0f;

    const size_t bn  = (size_t)b * N;
    const float* fi  = frac + (bn + i) * 3;
    const float fi0 = fi[0], fi1 = fi[1], fi2 = fi[2];
    const bool  ei  = ent[bn + i] != 0;
    const size_t row = (bn + i) * (size_t)N;

    const bool fast = (j0 + 4 <= N) && (((bn + j0) & 3) == 0) &&
                      (((row + j0) & 3) == 0);

    float fx[4], fy[4], fz[4];
    uint8_t ej[4];
    int cnt;
    if (fast) {
        const v4f* FJ = (const v4f*)(frac + (bn + j0) * 3);   // 16B aligned
        v4f A = FJ[0], Bv = FJ[1], Cv = FJ[2];
        fx[0] = A.x;  fy[0] = A.y;  fz[0] = A.z;
        fx[1] = A.w;  fy[1] = Bv.x; fz[1] = Bv.y;
        fx[2] = Bv.z; fy[2] = Bv.w; fz[2] = Cv.x;
        fx[3] = Cv.y; fy[3] = Cv.z; fz[3] = Cv.w;
        uint32_t e4 = *(const uint32_t*)(ent + bn + j0);
        ej[0] = e4 & 0xff; ej[1] = (e4 >> 8) & 0xff;
        ej[2] = (e4 >> 16) & 0xff; ej[3] = (e4 >> 24) & 0xff;
        cnt = 4;
    } else {
        cnt = N - j0; if (cnt > 4) cnt = 4;
        for (int k = 0; k < cnt; ++k) {
            const float* F = frac + (bn + j0 + k) * 3;
            fx[k] = F[0]; fy[k] = F[1]; fz[k] = F[2];
            ej[k] = ent[bn + j0 + k];
        }
    }

    float vx[4], vy[4], vz[4], ss[4], mm[4];
#pragma unroll
    for (int k = 0; k < 4; ++k) {
        if (k >= cnt) { vx[k]=vy[k]=vz[k]=ss[k]=mm[k]=0.0f; continue; }
        float d0 = fx[k] - fi0; d0 = fmaf(-rintf(d0), p0, d0);
        float d1 = fy[k] - fi1; d1 = fmaf(-rintf(d1), p1, d1);
        float d2 = fz[k] - fi2; d2 = fmaf(-rintf(d2), p2, d2);
        float x = fmaf(d2, c20, fmaf(d1, c10, d0 * c00));
        float y = fmaf(d2, c21, fmaf(d1, c11, d0 * c01));
        float z = fmaf(d2, c22, fmaf(d1, c12, d0 * c02));
        float s = fmaf(z, z, fmaf(y, y, x * x));
        int j = j0 + k;
        bool ok = ei && (ej[k] != 0) && (j != i) && (s < RC2);
        float m = ok ? 1.0f : 0.0f;
        vx[k] = x * m; vy[k] = y * m; vz[k] = z * m;
        ss[k] = s * m; mm[k] = m;
    }

    if (fast) {
        v4f s0 = { vx[0], vy[0], vz[0], vx[1] };
        v4f s1 = { vy[1], vz[1], vx[2], vy[2] };
        v4f s2 = { vz[2], vx[3], vy[3], vz[3] };
        v4f* OV = (v4f*)(ovec + (row + j0) * 3);              // 16B aligned
        __builtin_nontemporal_store(s0, OV + 0);
        __builtin_nontemporal_store(s1, OV + 1);
        __builtin_nontemporal_store(s2, OV + 2);
        v4f sv = { ss[0], ss[1], ss[2], ss[3] };
        v4f mv = { mm[0], mm[1], mm[2], mm[3] };
        __builtin_nontemporal_store(sv, (v4f*)(osod + row + j0));
        __builtin_nontemporal_store(mv, (v4f*)(omsk + row + j0));
    } else {
        for (int k = 0; k < cnt; ++k) {
            size_t p = row + j0 + k;
            __builtin_nontemporal_store(vx[k], ovec + p * 3 + 0);
            __builtin_nontemporal_store(vy[k], ovec + p * 3 + 1);
            __builtin_nontemporal_store(vz[k], ovec + p * 3 + 2);
            __builtin_nontemporal_store(ss[k], osod + p);
            __builtin_nontemporal_store(mm[k], omsk + p);
        }
    }
}

extern "C" void kernel_launch(void* const* d_in, const int* in_sizes, int n_in,
                              void* d_out, int out_size, void* d_ws, size_t ws_size,
                              hipStream_t stream) {
    const float*   pos = (const float*)d_in[0];    // [B,N,3] f32
    const float*   cel = (const float*)d_in[1];    // [B,3,3] f32
    const uint8_t* pbc = (const uint8_t*)d_in[2];  // [B,3] bool (1 byte)
    const uint8_t* ent = (const uint8_t*)d_in[3];  // [B,N] bool (1 byte)

    const int B     = in_sizes[1] / 9;
    const int total = in_sizes[0] / 3;             // B*N
    const int N     = total / B;

    // workspace: inv[B*9] at offset 0, frac[B*N*3] at float offset 256 (16B ok)
    float* inv  = (float*)d_ws;
    float* frac = (float*)d_ws + 256;

    inv3x3_kernel<<<1, 32, 0, stream>>>(cel, inv, B);

    const int tiles = (total + 15) / 16;           // 16 atoms per wave
    frac_wmma_kernel<<<(tiles + 7) / 8, 256, 0, stream>>>(pos, inv, frac, N, total);

    float* ovec = (float*)d_out;
    float* osod = ovec + (size_t)total * N * 3;
    float* omsk = osod + (size_t)total * N;

    dim3 grid((N + 4 * 256 - 1) / (4 * 256), N, B);
    pair_kernel<<<grid, 256, 0, stream>>>(frac, cel, pbc, ent,
                                          ovec, osod, omsk, N);
}